// self_attention_3633542333127
// MI455X (gfx1250) — compile-verified
//
#include <hip/hip_runtime.h>
#include <cmath>

// ---------------------------------------------------------------------------
// XCiT cross-covariance attention for MI455X (gfx1250), fp32 WMMA path.
// B=8, N=8192, C=384, H=8, Dh=48.  Memory-bound (~1.2 GB traffic @ 23.3 TB/s
// ~= 52us); fp32 V_WMMA_F32_16X16X4_F32 keeps reference precision.
// GEMM tiles are streamed with GLOBAL_LOAD_ASYNC_TO_LDS_B128 (ASYNCcnt) into
// double-buffered LDS so DRAM latency overlaps the WMMA pipe.
// ---------------------------------------------------------------------------

typedef float v2f __attribute__((ext_vector_type(2)));
typedef float v8f __attribute__((ext_vector_type(8)));

#define BATCH 8
#define SEQ   8192
#define CH    384
#define HEADS 8
#define DH    48

typedef __attribute__((address_space(3))) void lds_void;

static __device__ __forceinline__ v8f wmma_f32(v2f a, v2f b, v8f c) {
  // D = A(16x4,f32) * B(4x16,f32) + C(16x16,f32)
  return __builtin_amdgcn_wmma_f32_16x16x4_f32(
      /*neg_a=*/false, a, /*neg_b=*/false, b,
      /*c_mod=*/(short)0, c, /*reuse_a=*/false, /*reuse_b=*/false);
}

// Async 16B global->LDS copy (GV mode).  Tracked by ASYNCcnt.
static __device__ __forceinline__ void async_copy_b128(void* lds_dst,
                                                       const float* gsrc) {
  asm volatile("global_load_async_to_lds_b128 %0, %1, off"
               :: "v"((lds_void*)lds_dst), "v"(gsrc)
               : "memory");
}

static __device__ __forceinline__ void wait_async0() {
  asm volatile("s_wait_asynccnt 0x0" ::: "memory");
}

// ---------------------------------------------------------------------------
// Kernel 1: qkv = x @ w_qkv^T, written transposed into t[which][b][c][n]
// (c = h*Dh + d).  Block tile 128(M=bn) x 64(N=c'), 8 waves of 32x32.
// Double-buffered LDS tiles filled with async copies.
// ---------------------------------------------------------------------------
__global__ __launch_bounds__(256) void qkv_gemm_kernel(
    const float* __restrict__ x, const float* __restrict__ wqkv,
    float* __restrict__ t) {
  // stride 20 floats: 80B rows -> 16B aligned + conflict-free (m*20 mod 64
  // distinct for m=0..15).
  __shared__ float As[2][128][20];
  __shared__ float Bs[2][64][20];
  __shared__ float stage[64 * 128];

  const int tid  = threadIdx.x;
  const int lane = tid & 31;
  const int wave = tid >> 5;
  const int half = lane >> 4;   // 0 or 1
  const int l16  = lane & 15;
  const int mw   = wave >> 1;   // 0..3
  const int nw   = wave & 1;    // 0..1
  const int m0   = blockIdx.x * 128;  // global row index bn
  const int j0   = blockIdx.y * 64;   // global output column in [0, 3C)

  // per-thread tile-copy assignments
  const int ar0 = tid >> 2,        ag0 = (tid & 3) * 4;        // A row 0..63
  const int ar1 = (tid + 256) >> 2;                            // A row 64..127
  const int br  = tid >> 2,        bg  = (tid & 3) * 4;        // B row 0..63

  v8f acc[2][2] = {};

  // prologue: fill buffer 0
  async_copy_b128(&As[0][ar0][ag0], x + (size_t)(m0 + ar0) * CH + ag0);
  async_copy_b128(&As[0][ar1][ag0], x + (size_t)(m0 + ar1) * CH + ag0);
  async_copy_b128(&Bs[0][br][bg],   wqkv + (size_t)(j0 + br) * CH + bg);
  wait_async0();
  __syncthreads();

  for (int k0 = 0; k0 < CH; k0 += 16) {
    const int buf = (k0 >> 4) & 1;
    if (k0 + 16 < CH) {  // prefetch next chunk into the other buffer
      const int kn = k0 + 16;
      async_copy_b128(&As[buf ^ 1][ar0][ag0],
                      x + (size_t)(m0 + ar0) * CH + kn + ag0);
      async_copy_b128(&As[buf ^ 1][ar1][ag0],
                      x + (size_t)(m0 + ar1) * CH + kn + ag0);
      async_copy_b128(&Bs[buf ^ 1][br][bg],
                      wqkv + (size_t)(j0 + br) * CH + kn + bg);
    }
    const float (*Asb)[20] = As[buf];
    const float (*Bsb)[20] = Bs[buf];
#pragma unroll
    for (int ks = 0; ks < 4; ++ks) {
      v2f a[2], bf[2];
#pragma unroll
      for (int mi = 0; mi < 2; ++mi) {
        int m = mw * 32 + mi * 16 + l16;
        a[mi][0] = Asb[m][ks * 4 + 2 * half];
        a[mi][1] = Asb[m][ks * 4 + 2 * half + 1];
      }
#pragma unroll
      for (int ni = 0; ni < 2; ++ni) {
        int n = nw * 32 + ni * 16 + l16;
        bf[ni][0] = Bsb[n][ks * 4 + 2 * half];
        bf[ni][1] = Bsb[n][ks * 4 + 2 * half + 1];
      }
#pragma unroll
      for (int mi = 0; mi < 2; ++mi)
#pragma unroll
        for (int ni = 0; ni < 2; ++ni)
          acc[mi][ni] = wmma_f32(a[mi], bf[ni], acc[mi][ni]);
    }
    wait_async0();     // next-chunk copies (issued above) have landed
    __syncthreads();   // everyone done reading `buf` before it is refilled
  }

  // Stage transposed (stage[c_local][m_local]) so global writes are
  // contiguous along n.
#pragma unroll
  for (int mi = 0; mi < 2; ++mi)
#pragma unroll
    for (int ni = 0; ni < 2; ++ni)
#pragma unroll
      for (int v = 0; v < 8; ++v) {
        int ml = mw * 32 + mi * 16 + v + 8 * half;
        int cl = nw * 32 + ni * 16 + l16;
        stage[cl * 128 + ml] = acc[mi][ni][v];
      }
  __syncthreads();

  const int b  = m0 >> 13;          // m0 / SEQ  (SEQ = 8192)
  const int n0 = m0 & (SEQ - 1);
  for (int idx = tid; idx < 64 * 32; idx += 256) {
    int cl = idx >> 5, m4 = (idx & 31) * 4;
    int cg = j0 + cl;              // global column in [0, 3C)
    int which = cg / CH;           // tile never crosses a CH boundary (384%64==0)
    int c = cg - which * CH;
    float* dst = t + (((size_t)which * BATCH + b) * CH + c) * SEQ + n0 + m4;
    *(float4*)dst = *(const float4*)&stage[cl * 128 + m4];
  }
}

// ---------------------------------------------------------------------------
// Kernel 2: per-row inverse L2 norm of q (which=0) and k (which=1) planes.
// Row r in [0, 2*B*C), contiguous SEQ floats.  invn[r] = 1/max(||row||, eps)
// ---------------------------------------------------------------------------
__global__ __launch_bounds__(256) void rownorm_kernel(
    const float* __restrict__ t, float* __restrict__ invn) {
  __shared__ float red[256];
  const float* row = t + (size_t)blockIdx.x * SEQ;
  float s = 0.f;
  for (int i = threadIdx.x; i < SEQ; i += 256) { float v = row[i]; s += v * v; }
  red[threadIdx.x] = s;
  __syncthreads();
  for (int off = 128; off > 0; off >>= 1) {
    if (threadIdx.x < off) red[threadIdx.x] += red[threadIdx.x + off];
    __syncthreads();
  }
  if (threadIdx.x == 0)
    invn[blockIdx.x] = 1.0f / fmaxf(sqrtf(red[0]), 1e-12f);
}

// ---------------------------------------------------------------------------
// Kernel 3: one workgroup per (b,h).
//  Phase 1: attn(48x48) = q(48xN) . k(48xN)^T, K split across 8 waves,
//           partials combined via LDS ds_add_f32.
//  Then scale by inverse norms, row softmax.
//  Phase 2: out(48xN) = attn . v(48xN), written back into the q-plane
//           (each workgroup only touches its own 48 rows -> race free).
// ---------------------------------------------------------------------------
__global__ __launch_bounds__(256) void attention_kernel(
    float* __restrict__ t, const float* __restrict__ invn) {
  __shared__ float attn[DH * DH];

  const int tid  = threadIdx.x;
  const int lane = tid & 31;
  const int wave = tid >> 5;
  const int half = lane >> 4;
  const int l16  = lane & 15;
  const int b = blockIdx.x >> 3;
  const int h = blockIdx.x & 7;

  const float* qp = t + ((size_t)(0 * BATCH + b) * CH + h * DH) * SEQ;
  const float* kp = t + ((size_t)(1 * BATCH + b) * CH + h * DH) * SEQ;
  const float* vp = t + ((size_t)(2 * BATCH + b) * CH + h * DH) * SEQ;
  float*       op = t + ((size_t)(0 * BATCH + b) * CH + h * DH) * SEQ;

  for (int i = tid; i < DH * DH; i += 256) attn[i] = 0.f;
  __syncthreads();

  // ---- Phase 1: 48x48 = q . k^T, each wave covers 1024 of K=8192 ----
  {
    v8f acc[3][3] = {};
    const int kbase = wave * (SEQ / 8);
#pragma unroll 2
    for (int kk = 0; kk < SEQ / 8; kk += 4) {
      const int kg = kbase + kk + 2 * half;
      v2f a[3], bf[3];
#pragma unroll
      for (int mi = 0; mi < 3; ++mi)
        a[mi] = *(const v2f*)(qp + (size_t)(mi * 16 + l16) * SEQ + kg);
#pragma unroll
      for (int ni = 0; ni < 3; ++ni)
        bf[ni] = *(const v2f*)(kp + (size_t)(ni * 16 + l16) * SEQ + kg);
#pragma unroll
      for (int mi = 0; mi < 3; ++mi)
#pragma unroll
        for (int ni = 0; ni < 3; ++ni)
          acc[mi][ni] = wmma_f32(a[mi], bf[ni], acc[mi][ni]);
    }
#pragma unroll
    for (int mi = 0; mi < 3; ++mi)
#pragma unroll
      for (int ni = 0; ni < 3; ++ni)
#pragma unroll
        for (int v = 0; v < 8; ++v) {
          int m = mi * 16 + v + 8 * half;
          int e = ni * 16 + l16;
          atomicAdd(&attn[m * DH + e], acc[mi][ni][v]);
        }
  }
  __syncthreads();

  // ---- scale by 1/||q_m|| * 1/||k_e|| (folded L2 normalization) ----
  const float* invq = invn + b * CH + h * DH;
  const float* invk = invn + BATCH * CH + b * CH + h * DH;
  for (int i = tid; i < DH * DH; i += 256) {
    int m = i / DH, e = i - m * DH;
    attn[i] *= invq[m] * invk[e];
  }
  __syncthreads();

  // ---- softmax over each of the 48 rows ----
  if (tid < DH) {
    float* r = attn + tid * DH;
    float mx = r[0];
    for (int e = 1; e < DH; ++e) mx = fmaxf(mx, r[e]);
    float s = 0.f;
    for (int e = 0; e < DH; ++e) { float ev = __expf(r[e] - mx); r[e] = ev; s += ev; }
    float inv = 1.0f / s;
    for (int e = 0; e < DH; ++e) r[e] *= inv;
  }
  __syncthreads();

  // ---- Phase 2: out = attn(48x48) . v(48xN), per-wave 1024-wide slice ----
  const int nbase = wave * (SEQ / 8);
  for (int nt = 0; nt < (SEQ / 8) / 16; ++nt) {
    const int ng = nbase + nt * 16 + l16;
    v8f acc[3] = {};
#pragma unroll
    for (int ks = 0; ks < 12; ++ks) {
      const int kr = ks * 4 + 2 * half;
      v2f a[3];
#pragma unroll
      for (int mi = 0; mi < 3; ++mi) {
        int base = (mi * 16 + l16) * DH + kr;
        a[mi][0] = attn[base];
        a[mi][1] = attn[base + 1];
      }
      v2f bv;
      bv[0] = vp[(size_t)kr * SEQ + ng];
      bv[1] = vp[(size_t)(kr + 1) * SEQ + ng];
#pragma unroll
      for (int mi = 0; mi < 3; ++mi)
        acc[mi] = wmma_f32(a[mi], bv, acc[mi]);
    }
#pragma unroll
    for (int mi = 0; mi < 3; ++mi)
#pragma unroll
      for (int v = 0; v < 8; ++v)
        op[(size_t)(mi * 16 + v + 8 * half) * SEQ + ng] = acc[mi][v];
  }
}

// ---------------------------------------------------------------------------
// Kernel 4: out = A @ w_proj^T + b_proj, A[bn][c] read from the (B,C,N)
// attention-output layout.  A tile kept k-major in LDS (AsT[k][n]) so async
// copies are contiguous and fragment reads are transpose-free.
// ---------------------------------------------------------------------------
__global__ __launch_bounds__(256) void proj_gemm_kernel(
    const float* __restrict__ t, const float* __restrict__ wproj,
    const float* __restrict__ bproj, float* __restrict__ out) {
  __shared__ float AsT[2][16][128];   // [k][n], rows 512B -> aligned, no conflicts
  __shared__ float Bs[2][64][20];

  const int tid  = threadIdx.x;
  const int lane = tid & 31;
  const int wave = tid >> 5;
  const int half = lane >> 4;
  const int l16  = lane & 15;
  const int mw   = wave >> 1;
  const int nw   = wave & 1;
  const int m0   = blockIdx.x * 128;
  const int j0   = blockIdx.y * 64;
  const int b    = m0 >> 13;
  const int n0   = m0 & (SEQ - 1);

  // per-thread copy assignments: A tile 16 k-rows x 128 n (512 x 16B), B 64x16
  const int akr0 = tid >> 5,         an0 = (tid & 31) * 4;
  const int akr1 = (tid + 256) >> 5;
  const int br   = tid >> 2,         bg  = (tid & 3) * 4;

  const float* aBase = t + (size_t)b * CH * SEQ + n0;  // + (c)*SEQ + n

  v8f acc[2][2] = {};

  async_copy_b128(&AsT[0][akr0][an0], aBase + (size_t)akr0 * SEQ + an0);
  async_copy_b128(&AsT[0][akr1][an0], aBase + (size_t)akr1 * SEQ + an0);
  async_copy_b128(&Bs[0][br][bg],     wproj + (size_t)(j0 + br) * CH + bg);
  wait_async0();
  __syncthreads();

  for (int k0 = 0; k0 < CH; k0 += 16) {
    const int buf = (k0 >> 4) & 1;
    if (k0 + 16 < CH) {
      const int kn = k0 + 16;
      async_copy_b128(&AsT[buf ^ 1][akr0][an0],
                      aBase + (size_t)(kn + akr0) * SEQ + an0);
      async_copy_b128(&AsT[buf ^ 1][akr1][an0],
                      aBase + (size_t)(kn + akr1) * SEQ + an0);
      async_copy_b128(&Bs[buf ^ 1][br][bg],
                      wproj + (size_t)(j0 + br) * CH + kn + bg);
    }
    const float (*Ak)[128] = AsT[buf];
    const float (*Bsb)[20] = Bs[buf];
#pragma unroll
    for (int ks = 0; ks < 4; ++ks) {
      v2f a[2], bf[2];
#pragma unroll
      for (int mi = 0; mi < 2; ++mi) {
        int m = mw * 32 + mi * 16 + l16;
        a[mi][0] = Ak[ks * 4 + 2 * half][m];
        a[mi][1] = Ak[ks * 4 + 2 * half + 1][m];
      }
#pragma unroll
      for (int ni = 0; ni < 2; ++ni) {
        int n = nw * 32 + ni * 16 + l16;
        bf[ni][0] = Bsb[n][ks * 4 + 2 * half];
        bf[ni][1] = Bsb[n][ks * 4 + 2 * half + 1];
      }
#pragma unroll
      for (int mi = 0; mi < 2; ++mi)
#pragma unroll
        for (int ni = 0; ni < 2; ++ni)
          acc[mi][ni] = wmma_f32(a[mi], bf[ni], acc[mi][ni]);
    }
    wait_async0();
    __syncthreads();
  }

  // Epilogue: add bias, direct stores (16 lanes -> 64B contiguous per row).
#pragma unroll
  for (int ni = 0; ni < 2; ++ni) {
    int j = j0 + nw * 32 + ni * 16 + l16;
    float bias = bproj[j];
#pragma unroll
    for (int mi = 0; mi < 2; ++mi)
#pragma unroll
      for (int v = 0; v < 8; ++v) {
        int row = m0 + mw * 32 + mi * 16 + v + 8 * half;
        out[(size_t)row * CH + j] = acc[mi][ni][v] + bias;
      }
  }
}

// ---------------------------------------------------------------------------
extern "C" void kernel_launch(void* const* d_in, const int* in_sizes, int n_in,
                              void* d_out, int out_size, void* d_ws, size_t ws_size,
                              hipStream_t stream) {
  const float* x     = (const float*)d_in[0];  // (B, N, C)
  const float* wqkv  = (const float*)d_in[1];  // (3C, C)
  const float* wproj = (const float*)d_in[2];  // (C, C)
  const float* bproj = (const float*)d_in[3];  // (C,)
  float* out = (float*)d_out;                  // (B, N, C)

  float* t    = (float*)d_ws;                          // (3, B, C, N) fp32
  float* invn = t + (size_t)3 * BATCH * CH * SEQ;      // (2, B, C)

  // 1) qkv projection, transposed output
  qkv_gemm_kernel<<<dim3((BATCH * SEQ) / 128, (3 * CH) / 64), 256, 0, stream>>>(
      x, wqkv, t);
  // 2) inverse L2 norms of q/k rows
  rownorm_kernel<<<2 * BATCH * CH, 256, 0, stream>>>(t, invn);
  // 3) cross-covariance attention (writes result into q plane of t)
  attention_kernel<<<BATCH * HEADS, 256, 0, stream>>>(t, invn);
  // 4) output projection + bias
  proj_gemm_kernel<<<dim3((BATCH * SEQ) / 128, CH / 64), 256, 0, stream>>>(
      t, wproj, bproj, out);
}